// TransformerEncoderAdaLN8M_57002805952803
// MI455X (gfx1250) — compile-verified
//
#include <hip/hip_runtime.h>
#include <math.h>

// ---------------------------------------------------------------------------
// TransformerEncoderAdaLN forward for MI455X (gfx1250, wave32).
// All dense GEMMs + attention use V_WMMA_F32_16X16X4_F32 (exact fp32).
// ---------------------------------------------------------------------------

typedef float v2f __attribute__((ext_vector_type(2)));
typedef float v8f __attribute__((ext_vector_type(8)));

#define B_   16
#define L_   1024
#define MTOK 16384   // B*L
#define HID_ 256
#define HEADS_ 8
#define HD_  32
#define TED_ 1024

__device__ __forceinline__ v8f wmma4(v2f a, v2f b, v8f c) {
  // D = A(16x4, f32) x B(4x16, f32) + C(16x16, f32)
  return __builtin_amdgcn_wmma_f32_16x16x4_f32(
      /*neg_a=*/false, a, /*neg_b=*/false, b,
      /*c_mod=*/(short)0, c, /*reuse_a=*/false, /*reuse_b=*/false);
}

// ---------------------------------------------------------------------------
// Generic WMMA GEMM: out = epi( A(MxK,row-major) @ W(NxK,row-major)^T + bias )
// Block = 128 threads = 4 waves; wave computes 16(M) x 64(N); block tile 64x64.
// Fragment loads are direct global float2 loads (A rows / W rows), per the
// f32 16x16x4 VGPR layout (lane l: row l&15, K = ks + 2*(l>>4) .. +1).
// ACT: 0 none, 1 tanh, 2 exact gelu, 3 relu
// ---------------------------------------------------------------------------
template<int ACT, bool ADD2, bool FILM, bool GATE, bool RES>
__global__ void __launch_bounds__(128)
wmma_gemm(const float* __restrict__ A, const float* __restrict__ W,
          const float* __restrict__ bias, float* __restrict__ out,
          int K, int N, int outStride, int outColOff,
          const float* __restrict__ add2,                      // pos emb (L,N)
          const float* __restrict__ fshift, const float* __restrict__ fscale,
          int filmStride,
          const float* __restrict__ gate, int gateStride,
          const float* __restrict__ res, int resStride)
{
  const int lane = threadIdx.x & 31;
  const int wave = threadIdx.x >> 5;
  const int m    = lane & 15;
  const int hi   = lane >> 4;
  const int rowBase = blockIdx.y * 64 + wave * 16;
  const int colBase = blockIdx.x * 64;

  const float* Arow = A + (size_t)(rowBase + m) * K + 2 * hi;
  const float* Wrow = W + (size_t)(colBase + m) * K + 2 * hi;

  v8f acc[4];
  #pragma unroll
  for (int j = 0; j < 4; ++j) acc[j] = (v8f){0,0,0,0,0,0,0,0};

  #pragma unroll 4
  for (int ks = 0; ks < K; ks += 4) {
    v2f a  = *(const v2f*)(Arow + ks);
    v2f b0 = *(const v2f*)(Wrow + ks);
    v2f b1 = *(const v2f*)(Wrow + (size_t)16 * K + ks);
    v2f b2 = *(const v2f*)(Wrow + (size_t)32 * K + ks);
    v2f b3 = *(const v2f*)(Wrow + (size_t)48 * K + ks);
    acc[0] = wmma4(a, b0, acc[0]);
    acc[1] = wmma4(a, b1, acc[1]);
    acc[2] = wmma4(a, b2, acc[2]);
    acc[3] = wmma4(a, b3, acc[3]);
  }

  // C layout: VGPR r -> row (rowBase + r + 8*hi), col = colBase + j*16 + m
  #pragma unroll
  for (int j = 0; j < 4; ++j) {
    const int col = colBase + j * 16 + m;
    #pragma unroll
    for (int r = 0; r < 8; ++r) {
      const int row = rowBase + r + 8 * hi;
      const int b   = row >> 10;           // batch = row / L
      float v = acc[j][r] + bias[col];
      if (ADD2) v += add2[(size_t)(row & (L_ - 1)) * N + col];
      if (FILM) v = fscale[b * filmStride + col] * v + fshift[b * filmStride + col];
      if (ACT == 1) v = tanhf(v);
      else if (ACT == 2) v = 0.5f * v * (1.0f + erff(v * 0.70710678118654752f));
      else if (ACT == 3) v = fmaxf(v, 0.0f);
      if (GATE) v *= gate[b * gateStride + col];
      if (RES)  v += res[(size_t)row * resStride + col];
      out[(size_t)row * outStride + outColOff + col] = v;
    }
  }
}

// ---------------------------------------------------------------------------
// Fused flash attention: one wave per (batch, head, 16-row q tile).
// q from hn (cols [0,256) of qkv buf), k/v from h (cols [256,768)).
// S = QK^T via 8 WMMAs (HD=32), online softmax with 16-lane shfl reductions,
// P re-layout via 16x16 LDS tile, O += P·V via 8 WMMAs per k tile.
// ---------------------------------------------------------------------------
__global__ void __launch_bounds__(32)
attention_kernel(const float* __restrict__ qkv, float* __restrict__ attn_out)
{
  const int qt  = blockIdx.x;    // q tile 0..63
  const int hId = blockIdx.y;    // head 0..7
  const int b   = blockIdx.z;    // batch 0..15
  const int lane = threadIdx.x;
  const int m  = lane & 15;
  const int hi = lane >> 4;
  __shared__ float sP[16 * 16];

  const size_t base = (size_t)b * L_ * 768;
  const float* qrow = qkv + base + (size_t)(qt * 16 + m) * 768 + hId * HD_ + 2 * hi;

  v2f qa[8];
  #pragma unroll
  for (int s = 0; s < 8; ++s) qa[s] = *(const v2f*)(qrow + s * 4);

  float rowmax[8], rowsum[8];
  #pragma unroll
  for (int r = 0; r < 8; ++r) { rowmax[r] = -1e30f; rowsum[r] = 0.0f; }
  v8f o0 = (v8f){0,0,0,0,0,0,0,0};
  v8f o1 = (v8f){0,0,0,0,0,0,0,0};
  const float scale = 0.17677669529663687f;   // 1/sqrt(32)

  for (int kt = 0; kt < L_ / 16; ++kt) {
    const float* krow = qkv + base + (size_t)(kt * 16 + m) * 768 + HID_ + hId * HD_ + 2 * hi;
    v8f s8 = (v8f){0,0,0,0,0,0,0,0};
    #pragma unroll
    for (int st = 0; st < 8; ++st) {
      v2f kb = *(const v2f*)(krow + st * 4);
      s8 = wmma4(qa[st], kb, s8);
    }
    #pragma unroll
    for (int r = 0; r < 8; ++r) {
      float sv = s8[r] * scale;
      float mx = sv;
      mx = fmaxf(mx, __shfl_xor(mx, 1, 32));
      mx = fmaxf(mx, __shfl_xor(mx, 2, 32));
      mx = fmaxf(mx, __shfl_xor(mx, 4, 32));
      mx = fmaxf(mx, __shfl_xor(mx, 8, 32));
      float newm = fmaxf(rowmax[r], mx);
      float corr = expf(rowmax[r] - newm);
      float p    = expf(sv - newm);
      float ps = p;
      ps += __shfl_xor(ps, 1, 32);
      ps += __shfl_xor(ps, 2, 32);
      ps += __shfl_xor(ps, 4, 32);
      ps += __shfl_xor(ps, 8, 32);
      rowsum[r] = rowsum[r] * corr + ps;
      rowmax[r] = newm;
      o0[r] *= corr;
      o1[r] *= corr;
      s8[r] = p;
    }
    // C-layout -> LDS (row-major 16x16)
    #pragma unroll
    for (int r = 0; r < 8; ++r) sP[(r + 8 * hi) * 16 + m] = s8[r];
    __syncthreads();
    // O += P(16x16) * V(16x32)
    const float* vbase = qkv + base + (size_t)(kt * 16 + 2 * hi) * 768 + 2 * HID_ + hId * HD_;
    #pragma unroll
    for (int st = 0; st < 4; ++st) {
      v2f pa = { sP[m * 16 + st * 4 + 2 * hi], sP[m * 16 + st * 4 + 2 * hi + 1] };
      const float* v0 = vbase + (size_t)(st * 4) * 768;
      const float* v1 = v0 + 768;
      v2f vb0 = { v0[m],      v1[m]      };
      v2f vb1 = { v0[16 + m], v1[16 + m] };
      o0 = wmma4(pa, vb0, o0);
      o1 = wmma4(pa, vb1, o1);
    }
    __syncthreads();
  }

  #pragma unroll
  for (int r = 0; r < 8; ++r) {
    const int row = qt * 16 + r + 8 * hi;
    const float inv = 1.0f / rowsum[r];
    float* o = attn_out + ((size_t)(b * L_ + row)) * HID_ + hId * HD_;
    o[m]      = o0[r] * inv;
    o[16 + m] = o1[r] * inv;
  }
}

// ---------------------------------------------------------------------------
// AdaLN: hn[row] = (h-mean)/sqrt(var+eps) * scale[b] + shift[b]  (256 cols)
// ---------------------------------------------------------------------------
__global__ void __launch_bounds__(256)
adaln_kernel(const float* __restrict__ h, const float* __restrict__ coeff,
             int coeffStride, int shiftOff, int scaleOff, float* __restrict__ hn)
{
  const int row = blockIdx.x;
  const int tid = threadIdx.x;
  __shared__ float red[256];
  const float x = h[(size_t)row * HID_ + tid];
  red[tid] = x; __syncthreads();
  for (int s = 128; s > 0; s >>= 1) { if (tid < s) red[tid] += red[tid + s]; __syncthreads(); }
  const float mean = red[0] * (1.0f / HID_);
  __syncthreads();
  const float d = x - mean;
  red[tid] = d * d; __syncthreads();
  for (int s = 128; s > 0; s >>= 1) { if (tid < s) red[tid] += red[tid + s]; __syncthreads(); }
  const float var = red[0] * (1.0f / HID_);
  const int b = row >> 10;
  const float sc = coeff[b * coeffStride + scaleOff + tid];
  const float sh = coeff[b * coeffStride + shiftOff + tid];
  hn[(size_t)row * HID_ + tid] = d * rsqrtf(var + 1e-5f) * sc + sh;
}

// ---------------------------------------------------------------------------
// Tiny (M=16) GEMMs on VALU: out = act(A @ W^T + bias), optional silu on A.
// ACT: 0 none, 1 silu.
// ---------------------------------------------------------------------------
template<int ACT, bool ASILU>
__global__ void small_gemm(const float* __restrict__ A, const float* __restrict__ W,
                           const float* __restrict__ bias, float* __restrict__ out,
                           int M, int N, int K)
{
  const int idx = blockIdx.x * blockDim.x + threadIdx.x;
  if (idx >= M * N) return;
  const int mm = idx / N, nn = idx % N;
  float acc = bias[nn];
  const float* a = A + (size_t)mm * K;
  const float* w = W + (size_t)nn * K;
  for (int k = 0; k < K; ++k) {
    float av = a[k];
    if (ASILU) av = av / (1.0f + expf(-av));
    acc += av * w[k];
  }
  if (ACT == 1) acc = acc / (1.0f + expf(-acc));
  out[idx] = acc;
}

// In-place gate transform: x -> 1 + 1.5*tanh(x/1.5) over a column slice.
__global__ void gate_transform(float* __restrict__ p, int rows, int rowStride,
                               int colOff, int cols)
{
  const int idx = blockIdx.x * blockDim.x + threadIdx.x;
  if (idx >= rows * cols) return;
  const int r = idx / cols, c = idx % cols;
  float* q = p + (size_t)r * rowStride + colOff + c;
  *q = 1.0f + 1.5f * tanhf(*q * (1.0f / 1.5f));
}

// timestep_embedding: t_f[b][i] = cos(t_b * f_i), t_f[b][64+i] = sin(..)
__global__ void timestep_kernel(const float* __restrict__ t, float* __restrict__ tf)
{
  const int b = blockIdx.x, i = threadIdx.x;   // 64 threads
  const float freq = expf(-logf(10000.0f) * (float)i / 64.0f);
  const float arg = t[b] * freq;
  tf[b * 128 + i]      = cosf(arg);
  tf[b * 128 + 64 + i] = sinf(arg);
}

// ---------------------------------------------------------------------------
extern "C" void kernel_launch(void* const* d_in, const int* in_sizes, int n_in,
                              void* d_out, int out_size, void* d_ws, size_t ws_size,
                              hipStream_t stream) {
  (void)in_sizes; (void)n_in; (void)out_size; (void)ws_size;
  const float* x = (const float*)d_in[0];
  const float* t = (const float*)d_in[1];
  int pi = 2;
  auto nxt = [&]() { return (const float*)d_in[pi++]; };
  const float* inp_w1 = nxt(); const float* inp_b1 = nxt();
  const float* inp_w2 = nxt(); const float* inp_b2 = nxt();
  const float* te_w1  = nxt(); const float* te_b1  = nxt();
  const float* te_w2  = nxt(); const float* te_b2  = nxt();
  const float* pos_emb = nxt();
  const float* Lp[5][10];
  for (int i = 0; i < 5; ++i) for (int j = 0; j < 10; ++j) Lp[i][j] = nxt();
  const float* Hp[2][8];
  for (int i = 0; i < 2; ++i) for (int j = 0; j < 8; ++j) Hp[i][j] = nxt();

  float* ws = (float*)d_ws;
  size_t off = 0;
  auto alloc = [&](size_t n) { float* p = ws + off; off += n; return p; };
  float* t_f     = alloc(16 * 128);
  float* te1     = alloc(16 * 1024);
  float* t_embed = alloc(16 * 1024);
  float* vec     = alloc(16 * 1536);        // per-layer AdaLN coeffs (in-place)
  float* f1c     = alloc(16 * 1024);        // FiLM1 coeffs
  float* f2c     = alloc(16 * 256);         // FiLM2 coeffs
  float* h       = alloc((size_t)MTOK * 256);
  float* hn      = alloc((size_t)MTOK * 256);
  float* abuf    = alloc((size_t)MTOK * 256);   // inp-mlp mid / attn out
  float* big     = alloc((size_t)MTOK * 1024);  // qkv / ffn mid / film z (shared)
  float* outp    = (float*)d_out;

  const float* NP = nullptr;
  dim3 blk(128);

  // ---- time embedding path (tiny) ----
  timestep_kernel<<<16, 64, 0, stream>>>(t, t_f);
  small_gemm<1, false><<<(16 * 1024 + 255) / 256, 256, 0, stream>>>(t_f, te_w1, te_b1, te1, 16, 1024, 128);
  small_gemm<0, false><<<(16 * 1024 + 255) / 256, 256, 0, stream>>>(te1, te_w2, te_b2, t_embed, 16, 1024, 1024);

  // ---- input MLP + pos emb ----
  wmma_gemm<1, false, false, false, false><<<dim3(4, 256), blk, 0, stream>>>(
      x, inp_w1, inp_b1, abuf, 128, 256, 256, 0, NP, NP, NP, 0, NP, 0, NP, 0);
  wmma_gemm<0, true, false, false, false><<<dim3(4, 256), blk, 0, stream>>>(
      abuf, inp_w2, inp_b2, h, 256, 256, 256, 0, pos_emb, NP, NP, 0, NP, 0, NP, 0);

  // ---- transformer layers ----
  for (int l = 0; l < 5; ++l) {
    const float* qkv_w = Lp[l][0]; const float* qkv_b = Lp[l][1];
    const float* out_w = Lp[l][2]; const float* out_b = Lp[l][3];
    const float* ffn_w1 = Lp[l][4]; const float* ffn_b1 = Lp[l][5];
    const float* ffn_w2 = Lp[l][6]; const float* ffn_b2 = Lp[l][7];
    const float* tc_w = Lp[l][8]; const float* tc_b = Lp[l][9];

    // vec = silu(t_embed) @ tc_w^T + tc_b ; chunks b1,g1,b2,g2,a1,a2 (256 each)
    small_gemm<0, true><<<(16 * 1536 + 255) / 256, 256, 0, stream>>>(t_embed, tc_w, tc_b, vec, 16, 1536, 1024);
    gate_transform<<<(16 * 256 + 255) / 256, 256, 0, stream>>>(vec, 16, 1536, 256, 256);  // g1
    gate_transform<<<(16 * 256 + 255) / 256, 256, 0, stream>>>(vec, 16, 1536, 768, 256);  // g2

    // hn = adaLN(h; b1, g1)
    adaln_kernel<<<MTOK, 256, 0, stream>>>(h, vec, 1536, 0, 256, hn);

    // qkv buffer: q from hn, k/v from h  (qkv_w rows [0,256) / [256,768))
    wmma_gemm<0, false, false, false, false><<<dim3(4, 256), blk, 0, stream>>>(
        hn, qkv_w, qkv_b, big, 256, 256, 768, 0, NP, NP, NP, 0, NP, 0, NP, 0);
    wmma_gemm<0, false, false, false, false><<<dim3(8, 256), blk, 0, stream>>>(
        h, qkv_w + 256 * 256, qkv_b + 256, big, 256, 512, 768, 256, NP, NP, NP, 0, NP, 0, NP, 0);

    attention_kernel<<<dim3(64, 8, 16), 32, 0, stream>>>(big, abuf);

    // h = h + a1 * (abuf @ out_w^T + out_b)
    wmma_gemm<0, false, false, true, true><<<dim3(4, 256), blk, 0, stream>>>(
        abuf, out_w, out_b, h, 256, 256, 256, 0, NP, NP, NP, 0, vec + 4 * 256, 1536, h, 256);

    // hn = adaLN(h; b2, g2)
    adaln_kernel<<<MTOK, 256, 0, stream>>>(h, vec, 1536, 512, 768, hn);

    // ffn: big = gelu(hn @ w1^T + b1); h = h + a2 * (big @ w2^T + b2)
    wmma_gemm<2, false, false, false, false><<<dim3(16, 256), blk, 0, stream>>>(
        hn, ffn_w1, ffn_b1, big, 256, 1024, 1024, 0, NP, NP, NP, 0, NP, 0, NP, 0);
    wmma_gemm<0, false, false, true, true><<<dim3(4, 256), blk, 0, stream>>>(
        big, ffn_w2, ffn_b2, h, 1024, 256, 256, 0, NP, NP, NP, 0, vec + 5 * 256, 1536, h, 256);
  }

  // ---- FiLM heads (mu -> out cols [0,128), lv -> [128,256)) ----
  for (int hd = 0; hd < 2; ++hd) {
    const float* w1 = Hp[hd][0]; const float* b1 = Hp[hd][1];
    const float* w2 = Hp[hd][2]; const float* b2 = Hp[hd][3];
    const float* f1_w = Hp[hd][4]; const float* f1_b = Hp[hd][5];
    const float* f2_w = Hp[hd][6]; const float* f2_b = Hp[hd][7];

    small_gemm<0, false><<<(16 * 1024 + 255) / 256, 256, 0, stream>>>(t_embed, f1_w, f1_b, f1c, 16, 1024, 1024);
    gate_transform<<<(16 * 512 + 255) / 256, 256, 0, stream>>>(f1c, 16, 1024, 512, 512);
    small_gemm<0, false><<<(16 * 256 + 255) / 256, 256, 0, stream>>>(t_embed, f2_w, f2_b, f2c, 16, 256, 1024);
    gate_transform<<<(16 * 128 + 255) / 256, 256, 0, stream>>>(f2c, 16, 256, 128, 128);

    // z = relu(film1(h @ w1^T + b1))   (N=512)
    wmma_gemm<3, false, true, false, false><<<dim3(8, 256), blk, 0, stream>>>(
        h, w1, b1, big, 256, 512, 512, 0, NP, f1c, f1c + 512, 1024, NP, 0, NP, 0);
    // out[:, hd*128 : ] = film2(z @ w2^T + b2)   (N=128)
    wmma_gemm<0, false, true, false, false><<<dim3(2, 256), blk, 0, stream>>>(
        big, w2, b2, outp, 512, 128, 256, hd * 128, NP, f2c, f2c + 128, 256, NP, 0, NP, 0);
  }
}